// ChebConv2D_1666447310963
// MI455X (gfx1250) — compile-verified
//
#include <hip/hip_runtime.h>
#include <stdint.h>

// ---------------------------------------------------------------------------
// ChebConv2D for MI455X (gfx1250).
// Bandwidth-bound (3.6 GFLOP vs ~400MB traffic @ 23.3 TB/s) -> stay in f32,
// use V_WMMA_F32_16X16X4_F32 for the Chebyshev transforms, fuse each axis
// pass into one kernel, stage tiles through LDS, write the pass-1 result
// transposed (y_t[b,o,w,h]) so pass 2 is the *same* kernel reading
// contiguous rows.  Workspace: 16KB cheb matrices + 128MiB intermediate.
// ---------------------------------------------------------------------------

#ifndef __has_builtin
#define __has_builtin(x) 0
#endif

typedef float v2f __attribute__((ext_vector_type(2)));
typedef float v8f __attribute__((ext_vector_type(8)));
typedef int   v4i __attribute__((ext_vector_type(4)));

#define B_        32
#define CH        64      // in channels == out channels
#define HH        128
#define WW        128
#define K9        9       // degree+1
#define K16       16      // k padded to 16 for WMMA tiling
#define XS_STRIDE 132     // 128 + 4 pad: conflict-free b64 row reads
#define TK_STRIDE 20      // 16  + 4 pad

// ---- async global->LDS (CDNA5) with safe fallback --------------------------
#if __has_builtin(__builtin_amdgcn_global_load_async_to_lds_b128)
#define ASYNC_LDS 1
#endif

__device__ __forceinline__ void g2l_b128(const float* __restrict__ g, float* l) {
#if defined(ASYNC_LDS)
  // toolchain signature (from diagnostics): (as1 int4* src, as3 int4* dst,
  // int offset, int cpol)
  __builtin_amdgcn_global_load_async_to_lds_b128(
      (__attribute__((address_space(1))) v4i*)g,
      (__attribute__((address_space(3))) v4i*)l, 0, 0);
#else
  *(float4*)l = *(const float4*)g;
#endif
}

__device__ __forceinline__ void wait_async_lds() {
#if defined(ASYNC_LDS)
#if __has_builtin(__builtin_amdgcn_s_wait_asynccnt)
  __builtin_amdgcn_s_wait_asynccnt(0);
#else
  asm volatile("s_wait_asynccnt 0" ::: "memory");
#endif
#endif
}

// ---- f32 WMMA: D(16x16) = A(16x4) * B(4x16) + C ----------------------------
__device__ __forceinline__ v8f wmma_f32(v2f a, v2f b, v8f c) {
  // 8-arg pattern: (neg_a, A, neg_b, B, c_mod, C, reuse_a, reuse_b)
  return __builtin_amdgcn_wmma_f32_16x16x4_f32(false, a, false, b, (short)0, c,
                                               false, false);
}

// ---------------------------------------------------------------------------
// Kernel 0: build Tp16[16][128] (pinv rows, rows 9..15 zero) and
//           Te16[128][16] (T with cols 9..15 zero).  pinv(T)=(T^T T)^-1 T^T.
// ---------------------------------------------------------------------------
__global__ void cheb_precompute(float* __restrict__ tp16, float* __restrict__ te16) {
  __shared__ float  T[HH][K9];
  __shared__ double G[K9][2 * K9];
  const int t = threadIdx.x;

  if (t < HH) {
    double xi = (double)t / (double)(HH - 1);
    double xn = 2.0 * xi - 1.0;
    if (xn < -1.0 + 1e-6) xn = -1.0 + 1e-6;
    if (xn >  1.0 - 1e-6) xn =  1.0 - 1e-6;
    float th = acosf((float)xn);
    for (int k = 0; k < K9; ++k) T[t][k] = cosf(th * (float)k);
  }
  __syncthreads();

  if (t == 0) {
    for (int a = 0; a < K9; ++a)
      for (int b = 0; b < K9; ++b) {
        double s = 0.0;
        for (int i = 0; i < HH; ++i) s += (double)T[i][a] * (double)T[i][b];
        G[a][b] = s;
        G[a][K9 + b] = (a == b) ? 1.0 : 0.0;
      }
    // Gauss-Jordan with partial pivoting on [G | I]
    for (int col = 0; col < K9; ++col) {
      int piv = col;
      for (int r = col + 1; r < K9; ++r)
        if (fabs(G[r][col]) > fabs(G[piv][col])) piv = r;
      if (piv != col)
        for (int j = 0; j < 2 * K9; ++j) {
          double tmp = G[col][j]; G[col][j] = G[piv][j]; G[piv][j] = tmp;
        }
      double d = G[col][col];
      for (int j = 0; j < 2 * K9; ++j) G[col][j] /= d;
      for (int r = 0; r < K9; ++r)
        if (r != col) {
          double f = G[r][col];
          for (int j = 0; j < 2 * K9; ++j) G[r][j] -= f * G[col][j];
        }
    }
  }
  __syncthreads();

  if (t < HH) {
    for (int k = 0; k < K16; ++k) {
      float v = 0.f;
      if (k < K9) {
        double s = 0.0;
        for (int j = 0; j < K9; ++j) s += G[k][K9 + j] * (double)T[t][j];
        v = (float)s;
      }
      tp16[k * HH + t] = v;                            // [16][128]
      te16[t * K16 + k] = (k < K9) ? T[t][k] : 0.f;    // [128][16]
    }
  }
}

// ---------------------------------------------------------------------------
// Fused per-axis pass.  Both passes share index algebra:
//   in  row (c, fix):  in [((b*64+c)*128 + fix)*128 + 0..127]   (contiguous)
//   out elem (row,col): out[((b*64+row)*128 + col)*128 + fix]   (transposed)
// Pass1: in=x[b,c,h,w] fix=h, wgt=W_w, out=y_t[b,o,w,h]
// Pass2: in=y_t         fix=w, wgt=W_h, out=final [b,f,h,w]
// ---------------------------------------------------------------------------
__global__ __launch_bounds__(128) void cheb_stage(
    const float* __restrict__ in, const float* __restrict__ wgt,
    float* __restrict__ out, const float* __restrict__ tp16,
    const float* __restrict__ te16) {
  __shared__ float xs [CH  * XS_STRIDE];  // input tile  [64][128]
  __shared__ float tps[K16 * XS_STRIDE];  // Tp16        [16][128]
  __shared__ float tes[WW  * TK_STRIDE];  // Te16        [128][16]
  __shared__ float cm [CH  * TK_STRIDE];  // coeffs      [64][16]
  __shared__ float mm [CH  * TK_STRIDE];  // mixed       [64][16]

  const int tid = threadIdx.x;
  const int b   = blockIdx.x >> 7;
  const int fix = blockIdx.x & 127;

  // ---- stage LDS (async to LDS on CDNA5) ----
  for (int i = tid; i < CH * 32; i += 128) {           // 64 rows x 32 b128
    int c = i >> 5, j = (i & 31) << 2;
    g2l_b128(in + ((size_t)(b * CH + c) * HH + fix) * WW + j,
             &xs[c * XS_STRIDE + j]);
  }
  for (int i = tid; i < K16 * 32; i += 128) {
    int k = i >> 5, j = (i & 31) << 2;
    g2l_b128(tp16 + k * WW + j, &tps[k * XS_STRIDE + j]);
  }
  for (int i = tid; i < WW * 4; i += 128) {
    int w = i >> 2, j = (i & 3) << 2;
    g2l_b128(te16 + w * K16 + j, &tes[w * TK_STRIDE + j]);
  }
  wait_async_lds();
  __syncthreads();

  const int wv   = tid >> 5;            // wave = M-tile (16 channels)
  const int lane = tid & 31;
  const int lrow = lane & 15;           // l % 16
  const int lhi  = (lane >> 4) << 1;    // 2*(l/16)
  const int rbase = (lane >> 4) << 3;   // C/D row group

  // ---- stage 1: cm[64c][16k] = X[64x128] * Tp16^T[128x16]  (K=128) ----
  {
    v8f acc = {0.f, 0.f, 0.f, 0.f, 0.f, 0.f, 0.f, 0.f};
    const float* arow = &xs[(wv * 16 + lrow) * XS_STRIDE + lhi];
    const float* brow = &tps[lrow * XS_STRIDE + lhi];
#pragma unroll
    for (int kk = 0; kk < WW; kk += 4) {
      v2f a = {arow[kk], arow[kk + 1]};
      v2f bb = {brow[kk], brow[kk + 1]};
      acc = wmma_f32(a, bb, acc);
    }
#pragma unroll
    for (int r = 0; r < 8; ++r)
      cm[(wv * 16 + rbase + r) * TK_STRIDE + lrow] = acc[r];
  }
  __syncthreads();

  // ---- stage 2: per-degree channel mix  mm[o][k] = sum_c cm[c][k]*W[c,o,k]
  for (int idx = tid; idx < CH * K16; idx += 128) {
    int o = idx >> 4, k = idx & 15;
    float s = 0.f;
    if (k < K9) {
#pragma unroll 8
      for (int c = 0; c < CH; ++c)
        s += cm[c * TK_STRIDE + k] * wgt[(c * CH + o) * K9 + k];
    }
    mm[o * TK_STRIDE + k] = s;
  }
  __syncthreads();

  // ---- stage 3: out[64][128] = M[64x16] * Te16^T[16x128]  (K=16) ----
  {
    const float* arow = &mm[(wv * 16 + lrow) * TK_STRIDE + lhi];
#pragma unroll
    for (int nt = 0; nt < 8; ++nt) {
      v8f acc = {0.f, 0.f, 0.f, 0.f, 0.f, 0.f, 0.f, 0.f};
      const float* brow = &tes[(nt * 16 + lrow) * TK_STRIDE + lhi];
#pragma unroll
      for (int kk = 0; kk < K16; kk += 4) {
        v2f a = {arow[kk], arow[kk + 1]};
        v2f bb = {brow[kk], brow[kk + 1]};
        acc = wmma_f32(a, bb, acc);
      }
#pragma unroll
      for (int r = 0; r < 8; ++r) {
        int row = wv * 16 + rbase + r;     // output channel
        int col = nt * 16 + lrow;          // transformed-axis coordinate
        out[((size_t)(b * CH + row) * WW + col) * HH + fix] = acc[r];
      }
    }
  }
}

// ---------------------------------------------------------------------------
extern "C" void kernel_launch(void* const* d_in, const int* in_sizes, int n_in,
                              void* d_out, int out_size, void* d_ws,
                              size_t ws_size, hipStream_t stream) {
  (void)in_sizes; (void)n_in; (void)out_size; (void)ws_size;
  const float* x  = (const float*)d_in[0];   // [32,64,128,128]
  const float* Ww = (const float*)d_in[1];   // [64,64,9]
  const float* Wh = (const float*)d_in[2];   // [64,64,9]
  float* out = (float*)d_out;                // [32,64,128,128]

  float* tp16 = (float*)d_ws;                // 16*128 f32
  float* te16 = tp16 + K16 * HH;             // 128*16 f32
  float* yT   = te16 + HH * K16;             // 32*64*128*128 f32 (y transposed)

  cheb_precompute<<<1, 128, 0, stream>>>(tp16, te16);
  cheb_stage<<<B_ * HH, 128, 0, stream>>>(x,  Ww, yT,  tp16, te16);  // width pass
  cheb_stage<<<B_ * WW, 128, 0, stream>>>(yT, Wh, out, tp16, te16);  // height pass
}